// HTPLDA_11020886081787
// MI455X (gfx1250) — compile-verified
//
#include <hip/hip_runtime.h>
#include <hip/hip_bf16.h>

// ---------------------------------------------------------------------------
// HT-PLDA SGME scoring on gfx1250 (MI455X).
//   D=512, RANK=128, N=1024.  Output: [1024,1024] fp32 LLR matrix.
//
// Pipeline (all fp32, matching the reference precision):
//   Xc = (x - mu)^T                                [512 x 1024]
//   W  = H^T H             (fp32 WMMA 16x16x4)     [512 x 512]
//   P  = F^T W             (fp32 WMMA)             [128 x 512]
//   B0 = P F               (fp32 WMMA)             [128 x 128]
//   eigh(B0) -> lam, V     (parallel cyclic Jacobi, 128x129-padded LDS)
//   Q  = V^T P             (fp32 WMMA)             [128 x 512]
//   Z  = W Xc              (fp32 WMMA)             [512 x 1024]
//   wq_n = Xc_n . Z_n ;  q_n = wq_n - sum_r Y_rn^2/lam_r ; Y = Q Xc (WMMA)
//   b_n = (nu+384)/(nu+q_n) ; U_n = b_n Y_n ; l_n = 0.5(sum u^2/den - logdet)
//   S[t][e] = 0.5( sum_r (Ut+Ue)^2/(1+(bt+be)lam) - sum_r log(1+(bt+be)lam) )
//             - l_t - l_e         (VALU-bound: v_rcp_f32 + v_log_f32)
//
// CDNA5 paths: v_wmma_f32_16x16x4_f32 for all GEMMs; async global->LDS
// (GLOBAL_LOAD_ASYNC_TO_LDS_B128 + s_wait_asynccnt) for pairwise tile staging.
// ---------------------------------------------------------------------------

typedef float v2f __attribute__((ext_vector_type(2)));
typedef float v8f __attribute__((ext_vector_type(8)));
typedef int   v4i __attribute__((ext_vector_type(4)));

#define NCOL 1024
#define DDIM 512
#define RANK 128

#if __has_builtin(__builtin_amdgcn_global_load_async_to_lds_b128) && \
    __has_builtin(__builtin_amdgcn_s_wait_asynccnt)
#define USE_ASYNC_LDS 1
// Builtin signature (from hipcc diagnostic): param0 = v4i* in AS1 (printed as
// "__device__"), param1 = v4i* in AS3 (LDS), then imm offset, imm cpol.
typedef __attribute__((address_space(1))) v4i as1_v4i;
typedef __attribute__((address_space(3))) v4i as3_v4i;
#else
#define USE_ASYNC_LDS 0
#endif

// ---------------------------------------------------------------------------
// Generic fp32 WMMA GEMM: C[MxN] = op(A) * B, op(A)=A^T when ATRANS.
// One wave computes one 16x16 tile using V_WMMA_F32_16X16X4_F32.
// A 16x4 fp32 fragment: lane<16 holds row m=lane, v0=K(2*kh), v1=K(2*kh+1),
// kh = lane>>4.  B 4x16: v0 = rows {0,2} across lane halves, v1 = rows {1,3}.
// C/D: VGPR i -> row i (lanes 0-15) / row i+8 (lanes 16-31).
// ---------------------------------------------------------------------------
template<bool ATRANS>
__global__ __launch_bounds__(256)
void wmma_gemm_f32(const float* __restrict__ A, const float* __restrict__ B,
                   float* __restrict__ C, int M, int N, int K,
                   int lda, int ldb, int ldc)
{
    const int lane  = threadIdx.x & 31;
    const int wave  = threadIdx.x >> 5;
    const int tilesN = N >> 4;
    const int tile  = blockIdx.x * 8 + wave;          // wave-uniform
    if (tile >= (M >> 4) * tilesN) return;            // whole-wave exit (EXEC stays full)
    const int tm  = (tile / tilesN) << 4;
    const int tn  = (tile % tilesN) << 4;
    const int i15 = lane & 15;
    const int kh  = lane >> 4;                        // 0 or 1

    v8f acc = {0.f, 0.f, 0.f, 0.f, 0.f, 0.f, 0.f, 0.f};
    for (int k0 = 0; k0 < K; k0 += 4) {
        v2f a, b;
        if (ATRANS) {
            a.x = A[(k0 + 2 * kh + 0) * lda + (tm + i15)];
            a.y = A[(k0 + 2 * kh + 1) * lda + (tm + i15)];
        } else {
            // contiguous pair, 8B aligned -> single global_load_b64
            a = *(const v2f*)(A + (tm + i15) * lda + (k0 + 2 * kh));
        }
        b.x = B[(k0 + 2 * kh + 0) * ldb + (tn + i15)];
        b.y = B[(k0 + 2 * kh + 1) * ldb + (tn + i15)];
        acc = __builtin_amdgcn_wmma_f32_16x16x4_f32(
            /*neg_a=*/false, a, /*neg_b=*/false, b,
            /*c_mod=*/(short)0, acc, /*reuse_a=*/false, /*reuse_b=*/false);
    }
#pragma unroll
    for (int i = 0; i < 8; ++i)
        C[(tm + i + 8 * kh) * ldc + (tn + i15)] = acc[i];
}

// ---------------------------------------------------------------------------
// Xc[d][n] = x[n][d] - mu[d]
// ---------------------------------------------------------------------------
__global__ __launch_bounds__(256)
void center_kernel(const float* __restrict__ x, const float* __restrict__ mu,
                   float* __restrict__ Xc)
{
    int idx = blockIdx.x * 256 + threadIdx.x;   // over 512*1024
    int d = idx >> 10, n = idx & (NCOL - 1);
    Xc[idx] = x[n * DDIM + d] - mu[d];
}

// ---------------------------------------------------------------------------
// Parallel cyclic Jacobi eigensolver for symmetric 128x128 B0 in LDS.
// Round-robin (circle method) gives 64 disjoint rotations per round; a sweep
// is 127 rounds covering all pairs.  A and V live in LDS with row pitch 129
// floats to break 64-bank conflicts.  Eigen order/sign are irrelevant
// downstream (all consumers sum over r), so no sort is required.
// ---------------------------------------------------------------------------
__device__ __forceinline__ void pairPQ(int r, int i, int& p, int& q)
{
    if (i == 0) { p = 0; q = 1 + (r % 127); }
    else {
        p = 1 + ((r + i) % 127);
        q = 1 + ((r + 127 - i) % 127);
    }
}

#define JPITCH 129

__global__ __launch_bounds__(512)
void jacobi_eigh_128(const float* __restrict__ B0, float* __restrict__ Vout,
                     float* __restrict__ lamOut)
{
    extern __shared__ float sm[];
    float* A  = sm;                       // 128*129
    float* V  = sm + 128 * JPITCH;        // 128*129
    float* cr = V + 128 * JPITCH;         // 64 cosines
    float* sr = cr + 64;                  // 64 sines
    const int tid = threadIdx.x;

    for (int i = tid; i < 128 * 128; i += 512) {
        int r = i >> 7, c = i & 127;
        A[r * JPITCH + c] = 0.5f * (B0[r * 128 + c] + B0[c * 128 + r]); // symmetrize
        V[r * JPITCH + c] = (r == c) ? 1.0f : 0.0f;
    }
    __syncthreads();

    for (int sweep = 0; sweep < 12; ++sweep) {
        for (int round = 0; round < 127; ++round) {
            // Phase 1: rotation angles for the 64 disjoint pairs.
            if (tid < 64) {
                int p, q; pairPQ(round, tid, p, q);
                float app = A[p * JPITCH + p];
                float aqq = A[q * JPITCH + q];
                float apq = A[p * JPITCH + q];
                float c = 1.0f, s = 0.0f;
                if (fabsf(apq) > 1e-12f) {
                    float tau = (aqq - app) / (2.0f * apq);
                    float t = (tau >= 0.0f ? 1.0f : -1.0f) /
                              (fabsf(tau) + sqrtf(1.0f + tau * tau));
                    c = 1.0f / sqrtf(1.0f + t * t);
                    s = t * c;
                }
                cr[tid] = c; sr[tid] = s;
            }
            __syncthreads();
            // Phase 2: column rotations A <- A*J and V <- V*J (disjoint writes).
            for (int task = tid; task < 128 * 64; task += 512) {
                int row = task >> 6, i = task & 63;
                int p, q; pairPQ(round, i, p, q);
                float c = cr[i], s = sr[i];
                float ap = A[row * JPITCH + p], aq = A[row * JPITCH + q];
                A[row * JPITCH + p] = c * ap - s * aq;
                A[row * JPITCH + q] = s * ap + c * aq;
                float vp = V[row * JPITCH + p], vq = V[row * JPITCH + q];
                V[row * JPITCH + p] = c * vp - s * vq;
                V[row * JPITCH + q] = s * vp + c * vq;
            }
            __syncthreads();
            // Phase 3: row rotations A <- J^T*A.
            for (int task = tid; task < 128 * 64; task += 512) {
                int col = task >> 6, i = task & 63;
                int p, q; pairPQ(round, i, p, q);
                float c = cr[i], s = sr[i];
                float ap = A[p * JPITCH + col], aq = A[q * JPITCH + col];
                A[p * JPITCH + col] = c * ap - s * aq;
                A[q * JPITCH + col] = s * ap + c * aq;
            }
            __syncthreads();
        }
    }
    if (tid < 128) lamOut[tid] = A[tid * JPITCH + tid];
    for (int i = tid; i < 128 * 128; i += 512) {
        int r = i >> 7, c = i & 127;
        Vout[r * 128 + c] = V[r * JPITCH + c];
    }
}

// ---------------------------------------------------------------------------
// wq[n] = sum_d Xc[d][n] * Z[d][n]     (fully coalesced across n)
// ---------------------------------------------------------------------------
__global__ __launch_bounds__(256)
void colquad_kernel(const float* __restrict__ Xc, const float* __restrict__ Z,
                    float* __restrict__ wq)
{
    int n = blockIdx.x * 256 + threadIdx.x;
    float s = 0.0f;
    for (int d = 0; d < DDIM; ++d)
        s = fmaf(Xc[d * NCOL + n], Z[d * NCOL + n], s);
    wq[n] = s;
}

// ---------------------------------------------------------------------------
// Per-embedding stats: q, b, U (row-major [n][r]) and single-trial l_n.
// ---------------------------------------------------------------------------
__global__ __launch_bounds__(256)
void stats_kernel(const float* __restrict__ Y, const float* __restrict__ wq,
                  const float* __restrict__ lam, const float* __restrict__ nuPtr,
                  float* __restrict__ U, float* __restrict__ bOut,
                  float* __restrict__ lOut)
{
    __shared__ float lamS[RANK];
    if (threadIdx.x < RANK) lamS[threadIdx.x] = lam[threadIdx.x];
    __syncthreads();
    const int n = blockIdx.x * 256 + threadIdx.x;
    const float nu = nuPtr[0];

    float qs = 0.0f;
    for (int r = 0; r < RANK; ++r) {
        float y = Y[r * NCOL + n];
        qs += (y * y) / lamS[r];
    }
    float q = wq[n] - qs;
    float b = (nu + (float)(DDIM - RANK)) / (nu + q);

    float a1 = 0.0f, a2 = 0.0f;
    for (int r = 0; r < RANK; ++r) {
        float u = b * Y[r * NCOL + n];
        U[n * RANK + r] = u;
        float d = fmaf(b, lamS[r], 1.0f);
        a1 += (u * u) / d;
        a2 += __builtin_amdgcn_logf(d);          // v_log_f32 (base-2)
    }
    bOut[n] = b;
    lOut[n] = 0.5f * (a1 - 0.69314718055994531f * a2);
}

// ---------------------------------------------------------------------------
// Pairwise LLR tile kernel: 16(t) x 16(e) pairs per 256-thread workgroup.
// u-tiles staged in LDS with pitch 132 (bank-conflict free for both the
// ty-broadcast and tx-strided access patterns).  Tile staging goes through
// GLOBAL_LOAD_ASYNC_TO_LDS_B128 (cache->LDS DMA, ASYNCcnt) when available,
// bypassing VGPRs in the hottest kernel.  Inner loop is the hot path:
// v_fma + v_rcp_f32 + v_log_f32 per eigen-component.
// ---------------------------------------------------------------------------
#define UPITCH 132

__global__ __launch_bounds__(256)
void pairwise_kernel(const float* __restrict__ U, const float* __restrict__ bvec,
                     const float* __restrict__ lvec, const float* __restrict__ lam,
                     float* __restrict__ out)
{
    __shared__ float Ut[16 * UPITCH], Ue[16 * UPITCH], lamS[RANK];
    __shared__ float btS[16], beS[16], ltS[16], leS[16];
    const int tid = threadIdx.x;
    const int t0 = blockIdx.y << 4, e0 = blockIdx.x << 4;

    if (tid < RANK) lamS[tid] = lam[tid];
    if (tid >= 128 && tid < 144) {
        btS[tid - 128] = bvec[t0 + tid - 128];
        ltS[tid - 128] = lvec[t0 + tid - 128];
    }
    if (tid >= 144 && tid < 160) {
        beS[tid - 144] = bvec[e0 + tid - 144];
        leS[tid - 144] = lvec[e0 + tid - 144];
    }

#if USE_ASYNC_LDS
    // 16B granules: 512 per tile, 2 per thread per tile.  Both global and LDS
    // addresses are 16B aligned (row pitch 528B = 33*16B; U rows are 512B).
    {
#pragma unroll
        for (int j = 0; j < 2; ++j) {
            int g    = tid + j * 256;           // granule id within tile
            int row  = g >> 5;                  // 0..15
            int colf = (g & 31) * 4;            // float column, multiple of 4
            const float* gt = U + (size_t)(t0 + row) * RANK + colf;
            const float* ge = U + (size_t)(e0 + row) * RANK + colf;
            __builtin_amdgcn_global_load_async_to_lds_b128(
                (as1_v4i*)gt, (as3_v4i*)&Ut[row * UPITCH + colf], 0, 0);
            __builtin_amdgcn_global_load_async_to_lds_b128(
                (as1_v4i*)ge, (as3_v4i*)&Ue[row * UPITCH + colf], 0, 0);
        }
        __builtin_amdgcn_s_wait_asynccnt(0);    // drain this wave's async DMAs
    }
#else
    for (int i = tid; i < 16 * RANK; i += 256) {
        int row = i >> 7, r = i & 127;
        Ut[row * UPITCH + r] = U[(t0 + row) * RANK + r];
        Ue[row * UPITCH + r] = U[(e0 + row) * RANK + r];
    }
#endif
    __syncthreads();

    const int tx = tid & 15, ty = tid >> 4;
    const float bs = btS[ty] + beS[tx];
    const float* ut = &Ut[ty * UPITCH];
    const float* ue = &Ue[tx * UPITCH];

    float a1 = 0.0f, a2 = 0.0f;
#pragma unroll 4
    for (int r = 0; r < RANK; ++r) {
        float s = ut[r] + ue[r];
        float d = fmaf(bs, lamS[r], 1.0f);                 // >= 1, rcp/log safe
        a1 = fmaf(s * s, __builtin_amdgcn_rcpf(d), a1);    // v_rcp_f32
        a2 += __builtin_amdgcn_logf(d);                    // v_log_f32
    }
    out[(t0 + ty) * NCOL + (e0 + tx)] =
        0.5f * (a1 - 0.69314718055994531f * a2) - ltS[ty] - leS[tx];
}

// ---------------------------------------------------------------------------
// Host-side orchestration (all launches on `stream`, graph-capture safe).
// ---------------------------------------------------------------------------
extern "C" void kernel_launch(void* const* d_in, const int* in_sizes, int n_in,
                              void* d_out, int out_size, void* d_ws, size_t ws_size,
                              hipStream_t stream)
{
    const float* x  = (const float*)d_in[0];   // [1024, 512]
    const float* H  = (const float*)d_in[1];   // [512, 512]
    const float* F  = (const float*)d_in[2];   // [512, 128]
    const float* mu = (const float*)d_in[3];   // [512, 1]
    const float* nu = (const float*)d_in[4];   // scalar
    float* out = (float*)d_out;                // [1024, 1024]

    float* ws  = (float*)d_ws;                 // ~6.96 MB of fp32 scratch
    float* Xc  = ws;                  // 512*1024
    float* W   = Xc  + DDIM * NCOL;   // 512*512
    float* P   = W   + DDIM * DDIM;   // 128*512
    float* B0  = P   + RANK * DDIM;   // 128*128
    float* V   = B0  + RANK * RANK;   // 128*128
    float* lam = V   + RANK * RANK;   // 128
    float* Q   = lam + RANK;          // 128*512
    float* Z   = Q   + RANK * DDIM;   // 512*1024
    float* Y   = Z   + DDIM * NCOL;   // 128*1024
    float* wq  = Y   + RANK * NCOL;   // 1024
    float* bv  = wq  + NCOL;          // 1024
    float* lv  = bv  + NCOL;          // 1024
    float* U   = lv  + NCOL;          // 1024*128

    // 1) center
    center_kernel<<<(DDIM * NCOL) / 256, 256, 0, stream>>>(x, mu, Xc);
    // 2) W = H^T H                 (32x32 tiles = 1024 waves)
    wmma_gemm_f32<true ><<<(1024 + 7) / 8, 256, 0, stream>>>(H, H, W, 512, 512, 512, 512, 512, 512);
    // 3) P = F^T W                 (8x32 tiles)
    wmma_gemm_f32<true ><<<(256 + 7) / 8, 256, 0, stream>>>(F, W, P, 128, 512, 512, 128, 512, 512);
    // 4) B0 = P F                  (8x8 tiles)
    wmma_gemm_f32<false><<<(64 + 7) / 8, 256, 0, stream>>>(P, F, B0, 128, 128, 512, 512, 128, 128);
    // 5) eigh(B0) -> lam, V   (one WGP, 129-pitch A/V in 132.6 KB dynamic LDS)
    {
        size_t shm = (size_t)(2 * 128 * JPITCH + 128) * sizeof(float);
        jacobi_eigh_128<<<1, 512, shm, stream>>>(B0, V, lam);
    }
    // 6) Q = V^T P
    wmma_gemm_f32<true ><<<(256 + 7) / 8, 256, 0, stream>>>(V, P, Q, 128, 512, 128, 128, 512, 512);
    // 7) Z = W Xc                  (32x64 tiles)
    wmma_gemm_f32<false><<<(2048 + 7) / 8, 256, 0, stream>>>(W, Xc, Z, 512, 1024, 512, 512, 1024, 1024);
    // 8) wq[n] = Xc_n . Z_n
    colquad_kernel<<<NCOL / 256, 256, 0, stream>>>(Xc, Z, wq);
    // 9) Y = Q Xc                  (8x64 tiles)
    wmma_gemm_f32<false><<<(512 + 7) / 8, 256, 0, stream>>>(Q, Xc, Y, 128, 1024, 512, 512, 1024, 1024);
    // 10) per-embedding stats
    stats_kernel<<<NCOL / 256, 256, 0, stream>>>(Y, wq, lam, nu, U, bv, lv);
    // 11) pairwise LLR matrix
    dim3 g(NCOL / 16, NCOL / 16);
    pairwise_kernel<<<g, 256, 0, stream>>>(U, bv, lv, lam, out);
}